// ApproxEMD_82995948028479
// MI455X (gfx1250) — compile-verified
//
#include <hip/hip_runtime.h>
#include <hip/hip_bf16.h>

// Problem sizes (fixed by the reference setup).
#define Bsz 16
#define Nn  2048
#define Dd  256
#define BN  (Bsz * Nn)
#define NT  (Nn / 16)      // 128 column tiles per sweep
#define NBLK (Bsz * NT)    // 2048 one-wave blocks per sweep
#define EPSF 1e-9f

typedef __attribute__((ext_vector_type(2))) float v2f;
typedef __attribute__((ext_vector_type(8))) float v8f;

// ---- wave32 cross-lane reductions ------------------------------------------
__device__ __forceinline__ float rsum16(float v) {   // sum within each 16-lane half
  v += __shfl_xor(v, 1);
  v += __shfl_xor(v, 2);
  v += __shfl_xor(v, 4);
  v += __shfl_xor(v, 8);
  return v;
}
__device__ __forceinline__ float rsum32(float v) {   // full-wave sum
  v = rsum16(v);
  v += __shfl_xor(v, 16);
  return v;
}

// ---- 16x16 tile of rowsMat . colsMat^T over K=256 via V_WMMA_F32_16X16X4_F32
// aLds: LDS pointer to 16x256 row tile; bRow: global pointer to this lane's
// column row (colsMat[(c0+l15)*Dd]). Operand layout per CDNA5 ISA:
// lane&15 = M (A) / N (B); lane>>4 selects K pair {0,1} vs {2,3}.
__device__ __forceinline__ v8f gemm_tile16(const float* aP, const float* bP) {
  v8f g = {};
#pragma unroll 8
  for (int k = 0; k < Dd; k += 4) {
    v2f av = *(const v2f*)(aP + k);
    v2f bv = *(const v2f*)(bP + k);
    g = __builtin_amdgcn_wmma_f32_16x16x4_f32(false, av, false, bv,
                                              (short)0, g, false, false);
  }
  return g;
}

// Stage a contiguous 16x256 f32 row tile into LDS (one wave, float4 copies).
__device__ __forceinline__ void stage_tile(float* __restrict__ lds,
                                           const float* __restrict__ src,
                                           int lane) {
  const float4* s4 = (const float4*)src;
  float4* d4 = (float4*)lds;
#pragma unroll
  for (int j = 0; j < (16 * Dd / 4) / 32; ++j) d4[j * 32 + lane] = s4[j * 32 + lane];
  __syncthreads();
}

// ---- init: row norms, cost/currency = 1, zero the output scalar ------------
__global__ void init_kernel(const float* __restrict__ preds,
                            const float* __restrict__ labels,
                            float* __restrict__ pn, float* __restrict__ lm,
                            float* __restrict__ cost, float* __restrict__ cur,
                            float* __restrict__ out) {
  int i = blockIdx.x * blockDim.x + threadIdx.x;
  if (i >= BN) return;
  const float4* p = (const float4*)(preds + (size_t)i * Dd);
  const float4* l = (const float4*)(labels + (size_t)i * Dd);
  float sp = 0.f, sl = 0.f;
#pragma unroll 4
  for (int j = 0; j < Dd / 4; ++j) {
    float4 a = p[j]; sp += a.x * a.x + a.y * a.y + a.z * a.z + a.w * a.w;
    float4 c = l[j]; sl += c.x * c.x + c.y * c.y + c.z * c.z + c.w * c.w;
  }
  pn[i] = sp; lm[i] = sl; cost[i] = 1.f; cur[i] = 1.f;
  if (i == 0) out[0] = 0.f;
}

// ---- sweep 1: R[n] = sum_m exp(f*d[n,m]) * cost[m] -------------------------
__global__ __launch_bounds__(32)
void pass_rowsum(const float* __restrict__ P, const float* __restrict__ L,
                 const float* __restrict__ pn, const float* __restrict__ lm,
                 const float* __restrict__ cost, float* __restrict__ Rout,
                 float fac) {
  __shared__ __align__(16) float Atile[16 * Dd];
  int b = blockIdx.x / NT;
  int n0 = (blockIdx.x % NT) * 16;
  int lane = threadIdx.x, l15 = lane & 15, half = lane >> 4;

  stage_tile(Atile, P + ((size_t)b * Nn + n0) * Dd, lane);
  const float* aP = Atile + l15 * Dd + half * 2;

  int nbase = b * Nn + n0 + half * 8;
  float rn[8];
#pragma unroll
  for (int r = 0; r < 8; ++r) rn[r] = pn[nbase + r];

  float racc[8] = {0, 0, 0, 0, 0, 0, 0, 0};
  for (int mt = 0; mt < NT; ++mt) {
    int mcol = b * Nn + mt * 16 + l15;
    const float* bP = L + ((size_t)b * Nn + mt * 16 + l15) * Dd + half * 2;
    v8f g = gemm_tile16(aP, bP);
    float cn = lm[mcol];
    float cst = cost[mcol];
#pragma unroll
    for (int r = 0; r < 8; ++r) {
      float dist = rn[r] + cn - 2.0f * g[r];
      racc[r] += __expf(fac * dist) * cst;
    }
  }
#pragma unroll
  for (int r = 0; r < 8; ++r) racc[r] = rsum16(racc[r]);
  if (l15 == 0) {
#pragma unroll
    for (int r = 0; r < 8; ++r) Rout[nbase + r] = racc[r];
  }
}

// ---- W[n] = currency[n] / (R[n] + eps) -------------------------------------
__global__ void prep_w(const float* __restrict__ cur, const float* __restrict__ R,
                       float* __restrict__ W) {
  int i = blockIdx.x * blockDim.x + threadIdx.x;
  if (i < BN) W[i] = cur[i] / (R[i] + EPSF);
}

// ---- sweep 2 (transposed): C[m] = sum_n exp(f*d)*cost[m]*W[n] --------------
// Rows of the WMMA tile are LABEL indices (m), columns are PRED indices (n),
// so each wave owns 16 m's and writes C directly — no atomics.
__global__ __launch_bounds__(32)
void pass_colsum(const float* __restrict__ L, const float* __restrict__ P,
                 const float* __restrict__ lm, const float* __restrict__ pn,
                 const float* __restrict__ cost, const float* __restrict__ W,
                 float* __restrict__ Cout, float fac) {
  __shared__ __align__(16) float Atile[16 * Dd];
  int b = blockIdx.x / NT;
  int m0 = (blockIdx.x % NT) * 16;
  int lane = threadIdx.x, l15 = lane & 15, half = lane >> 4;

  stage_tile(Atile, L + ((size_t)b * Nn + m0) * Dd, lane);
  const float* aP = Atile + l15 * Dd + half * 2;

  int mbase = b * Nn + m0 + half * 8;
  float mn[8], cst[8];
#pragma unroll
  for (int r = 0; r < 8; ++r) { mn[r] = lm[mbase + r]; cst[r] = cost[mbase + r]; }

  float racc[8] = {0, 0, 0, 0, 0, 0, 0, 0};
  for (int nt = 0; nt < NT; ++nt) {
    int ncol = b * Nn + nt * 16 + l15;
    const float* bP = P + ((size_t)b * Nn + nt * 16 + l15) * Dd + half * 2;
    v8f g = gemm_tile16(aP, bP);
    float cn = pn[ncol];
    float wv = W[ncol];
#pragma unroll
    for (int r = 0; r < 8; ++r) {
      float dist = mn[r] + cn - 2.0f * g[r];
      racc[r] += __expf(fac * dist) * cst[r] * wv;
    }
  }
#pragma unroll
  for (int r = 0; r < 8; ++r) racc[r] = rsum16(racc[r]);
  if (l15 == 0) {
#pragma unroll
    for (int r = 0; r < 8; ++r) Cout[mbase + r] = racc[r];
  }
}

// ---- bid_wt[m] = min(cost[m]/(C[m]+eps), 1) --------------------------------
__global__ void prep_bidwt(const float* __restrict__ cost, const float* __restrict__ C,
                           float* __restrict__ bw) {
  int i = blockIdx.x * blockDim.x + threadIdx.x;
  if (i < BN) bw[i] = fminf(cost[i] / (C[i] + EPSF), 1.0f);
}

// ---- sweep 3: contribution + currency update -------------------------------
__global__ __launch_bounds__(32)
void pass_final(const float* __restrict__ P, const float* __restrict__ L,
                const float* __restrict__ pn, const float* __restrict__ lm,
                const float* __restrict__ cost, const float* __restrict__ W,
                const float* __restrict__ bw, float* __restrict__ cur,
                float* __restrict__ partial, float fac) {
  __shared__ __align__(16) float Atile[16 * Dd];
  int b = blockIdx.x / NT;
  int n0 = (blockIdx.x % NT) * 16;
  int lane = threadIdx.x, l15 = lane & 15, half = lane >> 4;

  stage_tile(Atile, P + ((size_t)b * Nn + n0) * Dd, lane);
  const float* aP = Atile + l15 * Dd + half * 2;

  int nbase = b * Nn + n0 + half * 8;
  float rn[8], wv[8];
#pragma unroll
  for (int r = 0; r < 8; ++r) { rn[r] = pn[nbase + r]; wv[r] = W[nbase + r]; }

  float racc[8] = {0, 0, 0, 0, 0, 0, 0, 0};
  float contrib = 0.f;
  for (int mt = 0; mt < NT; ++mt) {
    int mcol = b * Nn + mt * 16 + l15;
    const float* bP = L + ((size_t)b * Nn + mt * 16 + l15) * Dd + half * 2;
    v8f g = gemm_tile16(aP, bP);
    float cn = lm[mcol];
    float cst = cost[mcol];
    float bwm = bw[mcol];
#pragma unroll
    for (int r = 0; r < 8; ++r) {
      float dist = rn[r] + cn - 2.0f * g[r];
      float bf = __expf(fac * dist) * cst * wv[r] * bwm;
      contrib += bf * dist;
      racc[r] += bf;
    }
  }
#pragma unroll
  for (int r = 0; r < 8; ++r) racc[r] = rsum16(racc[r]);
  if (l15 == 0) {
#pragma unroll
    for (int r = 0; r < 8; ++r)
      cur[nbase + r] = fmaxf(cur[nbase + r] - racc[r], 0.0f);
  }
  contrib = rsum32(contrib);
  if (lane == 0) partial[blockIdx.x] = contrib;
}

// ---- deterministic partial reduction into the output scalar ----------------
__global__ void reduce_partials(const float* __restrict__ partial,
                                float* __restrict__ out) {
  __shared__ float s[256];
  int t = threadIdx.x;
  float v = 0.f;
  for (int i = t; i < NBLK; i += 256) v += partial[i];
  s[t] = v; __syncthreads();
  for (int o = 128; o > 0; o >>= 1) { if (t < o) s[t] += s[t + o]; __syncthreads(); }
  if (t == 0) out[0] += s[0];
}

// ---- cost[m] = max(cost[m] - C[m]*bid_wt[m], 0) ----------------------------
__global__ void update_cost(float* __restrict__ cost, const float* __restrict__ C,
                            const float* __restrict__ bw) {
  int i = blockIdx.x * blockDim.x + threadIdx.x;
  if (i < BN) cost[i] = fmaxf(cost[i] - C[i] * bw[i], 0.0f);
}

extern "C" void kernel_launch(void* const* d_in, const int* in_sizes, int n_in,
                              void* d_out, int out_size, void* d_ws, size_t ws_size,
                              hipStream_t stream) {
  (void)in_sizes; (void)n_in; (void)out_size; (void)ws_size;
  const float* preds  = (const float*)d_in[0];
  const float* labels = (const float*)d_in[1];
  float* out = (float*)d_out;

  // Workspace layout (~1.06 MB of the provided scratch).
  float* pn      = (float*)d_ws;          // [BN] pred row norms
  float* lm      = pn + BN;               // [BN] label row norms
  float* cost    = lm + BN;               // [BN]
  float* cur     = cost + BN;             // [BN] currency
  float* R       = cur + BN;              // [BN] row sums
  float* C       = R + BN;                // [BN] col sums
  float* W       = C + BN;                // [BN] currency/(R+eps)
  float* bw      = W + BN;                // [BN] bid weights
  float* partial = bw + BN;               // [NBLK] per-block scalar partials

  init_kernel<<<(BN + 255) / 256, 256, 0, stream>>>(preds, labels, pn, lm, cost, cur, out);

  // exp factors for i = 4,3,2,1,0,-1,-2 (last is 0.0)
  const float fac[7] = {-256.f, -64.f, -16.f, -4.f, -1.f, -0.25f, 0.f};
  for (int it = 0; it < 7; ++it) {
    float f = fac[it];
    pass_rowsum<<<NBLK, 32, 0, stream>>>(preds, labels, pn, lm, cost, R, f);
    prep_w<<<(BN + 255) / 256, 256, 0, stream>>>(cur, R, W);
    pass_colsum<<<NBLK, 32, 0, stream>>>(labels, preds, lm, pn, cost, W, C, f);
    prep_bidwt<<<(BN + 255) / 256, 256, 0, stream>>>(cost, C, bw);
    pass_final<<<NBLK, 32, 0, stream>>>(preds, labels, pn, lm, cost, W, bw, cur, partial, f);
    reduce_partials<<<1, 256, 0, stream>>>(partial, out);
    update_cost<<<(BN + 255) / 256, 256, 0, stream>>>(cost, C, bw);
  }
}